// MemoryGraph_35459249996528
// MI455X (gfx1250) — compile-verified
//
#include <hip/hip_runtime.h>
#include <hip/hip_bf16.h>

typedef __attribute__((ext_vector_type(2))) float v2f;
typedef __attribute__((ext_vector_type(8))) float v8f;

#define N_NODES 8192
#define KNBR 32
#define DDIM 64
#define DID 32
#define BSZ 4
#define TT 8
#define NODES_PER_WG 16
#define ROWS 64          // NODES_PER_WG * BSZ
#define XSTR 260         // padded LDS row stride (floats) to avoid bank conflicts
#define OSTR 68
#define NTHREADS 128

// ---------------------------------------------------------------------------
// Fused GEMM + bias + activation using V_WMMA_F32_16X16X4_F32.
// Computes Ydst[64 x Nout] = act(Xsrc[64 x K] @ Wg[K x Nout] + bias).
// 4 waves; wave w owns M-tile rows [16w, 16w+16). Weights are staged in LDS
// 128 K-rows x 16 N-cols at a time. ACT: 0=none, 1=silu, 2=tanh.
// ---------------------------------------------------------------------------
template<int ACT>
__device__ __forceinline__ void gemm_lds(
    const float* __restrict__ Wg, const float* __restrict__ bias,
    int K, int Nout,
    const float* Xsrc, float* Ydst, int ystr,
    float* Wt, int tid)
{
    const int lane  = tid & 31;
    const int r0    = (tid >> 5) << 4;      // wave's first row
    const int lrow  = lane & 15;            // A: M row / B,C: N col within tile
    const int khalf = (lane >> 4) << 1;     // 0 or 2 (K pair held by lane half)
    const int mofs  = (lane >> 4) << 3;     // 0 or 8 (C row offset)
    const int ntiles = (Nout + 15) >> 4;

    for (int nt = 0; nt < ntiles; ++nt) {
        v8f acc = {0.f, 0.f, 0.f, 0.f, 0.f, 0.f, 0.f, 0.f};
        for (int kb = 0; kb < K; kb += 128) {
            const int kl = (K - kb < 128) ? (K - kb) : 128;
            __syncthreads();
            for (int i = tid; i < (kl << 4); i += NTHREADS) {
                const int k = i >> 4, nl = i & 15;
                const int ncol = (nt << 4) + nl;
                Wt[i] = (ncol < Nout) ? Wg[(kb + k) * Nout + ncol] : 0.0f;
            }
            if ((kb + 128) < K)   // hint next K-block into cache while we compute
                __builtin_prefetch(&Wg[(kb + 128) * Nout + (nt << 4)], 0, 1);
            __syncthreads();
            const float* ap = Xsrc + (r0 + lrow) * XSTR + kb + khalf;
            const float* bp = Wt + khalf * 16 + lrow;
            for (int k0 = 0; k0 < kl; k0 += 4) {
                v2f a, b;
                a.x = ap[0]; a.y = ap[1];     // A 16x4 f32: lane half selects K pair
                b.x = bp[0]; b.y = bp[16];    // B 4x16 f32: vgpr selects K within half
                acc = __builtin_amdgcn_wmma_f32_16x16x4_f32(
                          false, a, false, b, (short)0, acc, false, false);
                ap += 4; bp += 64;
            }
        }
        const int ncol = (nt << 4) + lrow;
        if (ncol < Nout) {
            const float bv = bias[ncol];
            #pragma unroll
            for (int j = 0; j < 8; ++j) {     // C: vgpr j -> row j (+8 for hi lanes)
                float v = acc[j] + bv;
                if (ACT == 1)      v = v / (1.0f + __expf(-v));   // silu
                else if (ACT == 2) v = tanhf(v);
                Ydst[(r0 + j + mofs) * ystr + ncol] = v;
            }
        }
    }
    __syncthreads();
}

// ---------------------------------------------------------------------------
// One scan step. Workgroup = 16 nodes x 4 batch = 64 rows.
// Row r: node_local = r>>2, batch = r&3.
// ---------------------------------------------------------------------------
__global__ __launch_bounds__(NTHREADS, 1) void step_kernel(
    const float* __restrict__ cc,          // (BS,T,1024)
    const float* __restrict__ h_prev,      // (BS,N,64)
    const float* __restrict__ msgs_prev,   // (BS,N,64)
    const float* __restrict__ wconn_prev,  // (BS,N,32)
    const float* __restrict__ ident_prev,  // (N,32)
    const float* __restrict__ hebbian,     // (BS,N,32)
    const int*   __restrict__ conn,        // (N,32)
    const float* __restrict__ sw1, const float* __restrict__ sb1,
    const float* __restrict__ sw2, const float* __restrict__ sb2,
    const float* __restrict__ mw1, const float* __restrict__ mb1,
    const float* __restrict__ mw2, const float* __restrict__ mb2,
    const float* __restrict__ dw1, const float* __restrict__ db1,
    const float* __restrict__ dw2, const float* __restrict__ db2,
    float* __restrict__ h_next, float* __restrict__ msgs_next,
    float* __restrict__ wconn_next, float* __restrict__ ident_next,
    float* __restrict__ outp, int t)
{
    __shared__ float Xs[ROWS * XSTR];
    __shared__ float Ys[ROWS * XSTR];
    __shared__ float Os[ROWS * OSTR];
    __shared__ float Wt[128 * 16];
    __shared__ float decayS[ROWS];
    __shared__ float idnew[NODES_PER_WG * DID];

    const int tid = threadIdx.x;
    const int nodeBase = blockIdx.x * NODES_PER_WG;

    // ---- Phase 1: gather received + build mod_in = [heb|h|ident|recv|inject]
    {
        const int row  = tid >> 1;           // 0..63
        const int half = tid & 1;            // which 32 dims of D
        const int nl = row >> 2, b = row & 3;
        const int n = nodeBase + nl;
        float acc[32];
        #pragma unroll
        for (int j = 0; j < 32; ++j) acc[j] = 0.0f;
        const float* wrow = wconn_prev + ((size_t)b * N_NODES + n) * KNBR;
        const int*   crow = conn + (size_t)n * KNBR;
        for (int k = 0; k < KNBR; ++k) {
            const int idx = crow[k];
            const float w = 1.0f / (1.0f + __expf(-wrow[k]));   // sigmoid gate
            const float4* mp = (const float4*)(msgs_prev +
                               ((size_t)b * N_NODES + idx) * DDIM + half * 32);
            #pragma unroll
            for (int q = 0; q < 8; ++q) {
                const float4 v = mp[q];
                acc[q*4+0] += w * v.x; acc[q*4+1] += w * v.y;
                acc[q*4+2] += w * v.z; acc[q*4+3] += w * v.w;
            }
        }
        #pragma unroll
        for (int j = 0; j < 32; ++j) Xs[row * XSTR + 128 + half * 32 + j] = acc[j];
    }
    for (int i = tid; i < ROWS * KNBR; i += NTHREADS) {      // hebbian + ident cols
        const int r = i >> 5, j = i & 31;
        const int nl = r >> 2, b = r & 3, n = nodeBase + nl;
        Xs[r * XSTR + j]      = hebbian[((size_t)b * N_NODES + n) * KNBR + j];
        Xs[r * XSTR + 96 + j] = ident_prev[(size_t)n * DID + j];
    }
    for (int i = tid; i < ROWS * DDIM; i += NTHREADS) {      // h + inject cols
        const int r = i >> 6, d = i & 63;
        const int nl = r >> 2, b = r & 3, n = nodeBase + nl;
        Xs[r * XSTR + 32 + d]  = h_prev[((size_t)b * N_NODES + n) * DDIM + d];
        Xs[r * XSTR + 192 + d] = cc[((size_t)b * TT + t) * 1024 + (n >> 9) * DDIM + d];
    }
    __syncthreads();

    // ---- Phase 2: mod MLP (256 -> 256 silu -> 65)
    gemm_lds<1>(dw1, db1, 256, 256, Xs, Ys, XSTR, Wt, tid);
    gemm_lds<0>(dw2, db2, 256, 65,  Ys, Os, OSTR, Wt, tid);

    // ---- Phase 3: w_new, decay, ident_new (batch mean inside the WG)
    for (int i = tid; i < ROWS * KNBR; i += NTHREADS) {
        const int r = i >> 5, k = i & 31;
        const int nl = r >> 2, b = r & 3, n = nodeBase + nl;
        wconn_next[((size_t)b * N_NODES + n) * KNBR + k] = Os[r * OSTR + k];
    }
    if (tid < ROWS) decayS[tid] = 1.0f / (1.0f + __expf(-Os[tid * OSTR + KNBR]));
    for (int i = tid; i < NODES_PER_WG * DID; i += NTHREADS) {
        const int nl = i >> 5, j = i & 31, n = nodeBase + nl;
        float s = 0.0f;
        #pragma unroll
        for (int b = 0; b < BSZ; ++b) s += Os[(nl * 4 + b) * OSTR + 33 + j];
        const float v = ident_prev[(size_t)n * DID + j] + 0.25f * s;
        idnew[i] = v;
        ident_next[(size_t)n * DID + j] = v;
    }
    __syncthreads();

    // ---- Phase 4: state_in = [recv|inject|h|ide2] (224 cols) into Ys
    for (int i = tid; i < ROWS * DDIM; i += NTHREADS) {
        const int r = i >> 6, d = i & 63;
        Ys[r * XSTR + d]       = Xs[r * XSTR + 128 + d];
        Ys[r * XSTR + 64 + d]  = Xs[r * XSTR + 192 + d];
        Ys[r * XSTR + 128 + d] = Xs[r * XSTR + 32 + d];   // h_prev kept here
    }
    for (int i = tid; i < ROWS * DID; i += NTHREADS) {
        const int r = i >> 5, j = i & 31;
        Ys[r * XSTR + 192 + j] = idnew[(r >> 2) * DID + j];
    }
    __syncthreads();

    // ---- Phase 5: state MLP (224 -> 256 silu -> 64 tanh)
    gemm_lds<1>(sw1, sb1, 224, 256, Ys, Xs, XSTR, Wt, tid);
    gemm_lds<2>(sw2, sb2, 256, 64,  Xs, Os, OSTR, Wt, tid);

    // ---- Phase 6: h_new = decay*h + (1-decay)*tanh(...); emit output; msg_in
    for (int i = tid; i < ROWS * DDIM; i += NTHREADS) {
        const int r = i >> 6, d = i & 63;
        const int nl = r >> 2, b = r & 3, n = nodeBase + nl;
        const float hp  = Ys[r * XSTR + 128 + d];
        const float dec = decayS[r];
        const float hn  = dec * hp + (1.0f - dec) * Os[r * OSTR + d];
        h_next[((size_t)b * N_NODES + n) * DDIM + d] = hn;
        outp[(((size_t)b * TT + t) * N_NODES + n) * DDIM + d] = hn;  // word_states
        Ys[r * XSTR + d] = hn;                 // msg_in[:,0:64]
    }
    for (int i = tid; i < ROWS * DID; i += NTHREADS) {
        const int r = i >> 5, j = i & 31;
        Ys[r * XSTR + 64 + j] = idnew[(r >> 2) * DID + j];   // msg_in[:,64:96]
    }
    __syncthreads();

    // ---- Phase 7: msg MLP (96 -> 256 silu -> 64 tanh)
    gemm_lds<1>(mw1, mb1, 96,  256, Ys, Xs, XSTR, Wt, tid);
    gemm_lds<2>(mw2, mb2, 256, 64,  Xs, Os, OSTR, Wt, tid);

    // ---- Phase 8: msgs_next
    for (int i = tid; i < ROWS * DDIM; i += NTHREADS) {
        const int r = i >> 6, d = i & 63;
        const int nl = r >> 2, b = r & 3, n = nodeBase + nl;
        msgs_next[((size_t)b * N_NODES + n) * DDIM + d] = Os[r * OSTR + d];
    }
}

extern "C" void kernel_launch(void* const* d_in, const int* in_sizes, int n_in,
                              void* d_out, int out_size, void* d_ws, size_t ws_size,
                              hipStream_t stream) {
    (void)in_sizes; (void)n_in; (void)out_size; (void)ws_size;
    const float* cc   = (const float*)d_in[0];
    const float* h0   = (const float*)d_in[1];
    const float* m0   = (const float*)d_in[2];
    const float* w0   = (const float*)d_in[3];
    const float* heb  = (const float*)d_in[4];
    const float* id0  = (const float*)d_in[5];
    const float* sw1  = (const float*)d_in[6];
    const float* sb1  = (const float*)d_in[7];
    const float* sw2  = (const float*)d_in[8];
    const float* sb2  = (const float*)d_in[9];
    const float* mw1  = (const float*)d_in[10];
    const float* mb1  = (const float*)d_in[11];
    const float* mw2  = (const float*)d_in[12];
    const float* mb2  = (const float*)d_in[13];
    const float* dw1  = (const float*)d_in[14];
    const float* db1  = (const float*)d_in[15];
    const float* dw2  = (const float*)d_in[16];
    const float* db2  = (const float*)d_in[17];
    const int*   conn = (const int*)d_in[18];
    float* outp = (float*)d_out;

    float* ws = (float*)d_ws;
    const size_t HS = (size_t)BSZ * N_NODES * DDIM;   // 2,097,152 floats
    const size_t WSZ = (size_t)BSZ * N_NODES * KNBR;  // 1,048,576 floats
    const size_t IS = (size_t)N_NODES * DID;          //   262,144 floats
    float* hbuf[2] = { ws,            ws + HS };
    float* mbuf[2] = { ws + 2 * HS,   ws + 3 * HS };
    float* wbuf[2] = { ws + 4 * HS,   ws + 4 * HS + WSZ };
    float* ibuf[2] = { ws + 4 * HS + 2 * WSZ, ws + 4 * HS + 2 * WSZ + IS };

    for (int t = 0; t < TT; ++t) {
        const int cur = t & 1;
        const float* hp = (t == 0) ? h0  : hbuf[cur ^ 1];
        const float* mp = (t == 0) ? m0  : mbuf[cur ^ 1];
        const float* wp = (t == 0) ? w0  : wbuf[cur ^ 1];
        const float* ip = (t == 0) ? id0 : ibuf[cur ^ 1];
        step_kernel<<<N_NODES / NODES_PER_WG, NTHREADS, 0, stream>>>(
            cc, hp, mp, wp, ip, heb, conn,
            sw1, sb1, sw2, sb2, mw1, mb1, mw2, mb2, dw1, db1, dw2, db2,
            hbuf[cur], mbuf[cur], wbuf[cur], ibuf[cur], outp, t);
    }
}